// AllTripletLoss_13108240188033
// MI455X (gfx1250) — compile-verified
//
#include <hip/hip_runtime.h>

typedef __attribute__((ext_vector_type(2))) float v2f;
typedef __attribute__((ext_vector_type(8))) float v8f;

#define MARGIN    0.2f
#define NEG_FLOOR 0.6f
#define DDIM      128
#define NMAX      8192
#define NSLICE    8      // col-slices per row-panel
#define LDSS      132    // padded LDS row stride (floats) -> conflict-free b64 reads

// ---------------------------------------------------------------------------
// Phase 1: partial per-row max_pos / max_neg over this block's col slice.
// Block = 128 rows (8 waves x 16-row WMMA tile) x 64 col tiles of one slice.
// B tile staged global->regs->LDS, double buffered; all 8 waves share it.
// ---------------------------------------------------------------------------
__global__ __launch_bounds__(256) void triplet_phase1(
    const float* __restrict__ X, const int* __restrict__ tgt,
    float* __restrict__ mpP, float* __restrict__ mnP, int n) {
  __shared__ int   s_tgt[NMAX];
  __shared__ float s_b[2][16 * LDSS];

  const int tid = threadIdx.x;
  for (int i = tid; i < n; i += 256) s_tgt[i] = tgt[i];
  __syncthreads();

  const int lane  = tid & 31;
  const int wave  = tid >> 5;
  const int lo    = lane & 15;
  const int hi    = lane >> 4;
  const int panel = blockIdx.x / NSLICE;
  const int slice = blockIdx.x % NSLICE;
  const int rowBase = panel * 128 + wave * 16;
  const int nct   = n >> 4;
  const int tiles = nct / NSLICE;
  const int ct0   = slice * tiles;

  // A fragments: lane holds X[rowBase+lo][kb*4 + 2*hi + {0,1}]
  v2f a[32];
  const float* arow = X + (size_t)(rowBase + lo) * DDIM + hi * 2;
#pragma unroll
  for (int kb = 0; kb < 32; ++kb)
    a[kb] = *reinterpret_cast<const v2f*>(arow + kb * 4);

  int trow[8];
#pragma unroll
  for (int c = 0; c < 8; ++c) trow[c] = s_tgt[rowBase + c + 8 * hi];

  float mp[8], mn[8];
#pragma unroll
  for (int c = 0; c < 8; ++c) { mp[c] = -INFINITY; mn[c] = -INFINITY; }

  // staging role: thread moves 8 floats of the 16x128 B panel
  const int srow = tid >> 4;          // 0..15  (row within col tile)
  const int scol = (tid & 15) * 8;    // 0..120 (k offset)
  float4 r0, r1;
  {
    const float* p = X + (size_t)(ct0 * 16 + srow) * DDIM + scol;
    r0 = *reinterpret_cast<const float4*>(p);
    r1 = *reinterpret_cast<const float4*>(p + 4);
  }

  for (int i = 0; i < tiles; ++i) {
    const int ct  = ct0 + i;
    const int buf = i & 1;

    float* dst = &s_b[buf][srow * LDSS + scol];
    *reinterpret_cast<float4*>(dst)     = r0;
    *reinterpret_cast<float4*>(dst + 4) = r1;
    __syncthreads();

    // prefetch next tile's panel while computing this one
    if (i + 1 < tiles) {
      const float* p = X + (size_t)((ct + 1) * 16 + srow) * DDIM + scol;
      r0 = *reinterpret_cast<const float4*>(p);
      r1 = *reinterpret_cast<const float4*>(p + 4);
    }

    // compute 16x16 tile from LDS fragments (bank-conflict-free)
    const float* bbase = &s_b[buf][lo * LDSS + hi * 2];
    v8f acc = {};
#pragma unroll
    for (int kb = 0; kb < 32; ++kb) {
      v2f b = *reinterpret_cast<const v2f*>(bbase + kb * 4);
      acc = __builtin_amdgcn_wmma_f32_16x16x4_f32(false, a[kb], false, b,
                                                  (short)0, acc, false, false);
    }

    const int colBase = ct * 16;
    const int tcol = s_tgt[colBase + lo];
#pragma unroll
    for (int c = 0; c < 8; ++c) {
      const int r = rowBase + c + 8 * hi;
      const float s = acc[c];
      const bool same = (trow[c] == tcol);
      const bool diag = (r == colBase + lo);
      if (same && !diag) mp[c] = fmaxf(mp[c], s);
      if (!same)         mn[c] = fmaxf(mn[c], s);
    }
  }

  // reduce over the 16 lanes (same hi) holding one row's columns
#pragma unroll
  for (int c = 0; c < 8; ++c) {
#pragma unroll
    for (int off = 1; off < 16; off <<= 1) {
      mp[c] = fmaxf(mp[c], __shfl_xor(mp[c], off, 32));
      mn[c] = fmaxf(mn[c], __shfl_xor(mn[c], off, 32));
    }
  }
  if (lo == 0) {  // lanes 0 and 16 write this wave's 16 rows (disjoint per wave)
#pragma unroll
    for (int c = 0; c < 8; ++c) {
      const int r = rowBase + c + 8 * hi;
      mpP[(size_t)slice * n + r] = mp[c];
      mnP[(size_t)slice * n + r] = mn[c];
    }
  }
}

// merge slice partial maxima -> per-row max_pos / max_neg (fixed order)
__global__ void combine_max(const float* __restrict__ mpP, const float* __restrict__ mnP,
                            float* __restrict__ maxpos, float* __restrict__ maxneg, int n) {
  const int r = blockIdx.x * 256 + threadIdx.x;
  if (r < n) {
    float vp = -INFINITY, vn = -INFINITY;
    for (int s = 0; s < NSLICE; ++s) {
      vp = fmaxf(vp, mpP[(size_t)s * n + r]);
      vn = fmaxf(vn, mnP[(size_t)s * n + r]);
    }
    maxpos[r] = vp;
    maxneg[r] = vn;
  }
}

// ---------------------------------------------------------------------------
// Phase 2: thresholded partial sums per (slice, row)
// ---------------------------------------------------------------------------
__global__ __launch_bounds__(256) void triplet_phase2(
    const float* __restrict__ X, const int* __restrict__ tgt,
    const float* __restrict__ maxpos, const float* __restrict__ maxneg,
    float* __restrict__ psP, float* __restrict__ nsP, int* __restrict__ anP, int n) {
  __shared__ int   s_tgt[NMAX];
  __shared__ float s_b[2][16 * LDSS];

  const int tid = threadIdx.x;
  for (int i = tid; i < n; i += 256) s_tgt[i] = tgt[i];
  __syncthreads();

  const int lane  = tid & 31;
  const int wave  = tid >> 5;
  const int lo    = lane & 15;
  const int hi    = lane >> 4;
  const int panel = blockIdx.x / NSLICE;
  const int slice = blockIdx.x % NSLICE;
  const int rowBase = panel * 128 + wave * 16;
  const int nct   = n >> 4;
  const int tiles = nct / NSLICE;
  const int ct0   = slice * tiles;

  v2f a[32];
  const float* arow = X + (size_t)(rowBase + lo) * DDIM + hi * 2;
#pragma unroll
  for (int kb = 0; kb < 32; ++kb)
    a[kb] = *reinterpret_cast<const v2f*>(arow + kb * 4);

  int trow[8];
  float negth[8], posth[8];
#pragma unroll
  for (int c = 0; c < 8; ++c) {
    const int r = rowBase + c + 8 * hi;
    trow[c]  = s_tgt[r];
    negth[c] = maxneg[r] + MARGIN;                   // pos selected if sim < this
    posth[c] = fmaxf(NEG_FLOOR, maxpos[r]) - MARGIN; // neg selected if sim > this
  }

  float ps[8], ns[8];
  int   an[8];
#pragma unroll
  for (int c = 0; c < 8; ++c) { ps[c] = 0.f; ns[c] = 0.f; an[c] = 0; }

  const int srow = tid >> 4;
  const int scol = (tid & 15) * 8;
  float4 r0, r1;
  {
    const float* p = X + (size_t)(ct0 * 16 + srow) * DDIM + scol;
    r0 = *reinterpret_cast<const float4*>(p);
    r1 = *reinterpret_cast<const float4*>(p + 4);
  }

  for (int i = 0; i < tiles; ++i) {
    const int ct  = ct0 + i;
    const int buf = i & 1;

    float* dst = &s_b[buf][srow * LDSS + scol];
    *reinterpret_cast<float4*>(dst)     = r0;
    *reinterpret_cast<float4*>(dst + 4) = r1;
    __syncthreads();

    if (i + 1 < tiles) {
      const float* p = X + (size_t)((ct + 1) * 16 + srow) * DDIM + scol;
      r0 = *reinterpret_cast<const float4*>(p);
      r1 = *reinterpret_cast<const float4*>(p + 4);
    }

    const float* bbase = &s_b[buf][lo * LDSS + hi * 2];
    v8f acc = {};
#pragma unroll
    for (int kb = 0; kb < 32; ++kb) {
      v2f b = *reinterpret_cast<const v2f*>(bbase + kb * 4);
      acc = __builtin_amdgcn_wmma_f32_16x16x4_f32(false, a[kb], false, b,
                                                  (short)0, acc, false, false);
    }

    const int colBase = ct * 16;
    const int tcol = s_tgt[colBase + lo];
#pragma unroll
    for (int c = 0; c < 8; ++c) {
      const int r = rowBase + c + 8 * hi;
      const float s = acc[c];
      const bool same = (trow[c] == tcol);
      const bool diag = (r == colBase + lo);
      if (same && !diag && s < negth[c]) ps[c] += 1.0f - s;
      if (!same && s > posth[c]) { ns[c] += s; an[c] = 1; }
    }
  }

#pragma unroll
  for (int c = 0; c < 8; ++c) {
#pragma unroll
    for (int off = 1; off < 16; off <<= 1) {
      ps[c] += __shfl_xor(ps[c], off, 32);
      ns[c] += __shfl_xor(ns[c], off, 32);
      an[c] |= __shfl_xor(an[c], off, 32);
    }
  }
  if (lo == 0) {
#pragma unroll
    for (int c = 0; c < 8; ++c) {
      const int r = rowBase + c + 8 * hi;
      psP[(size_t)slice * n + r] = ps[c];
      nsP[(size_t)slice * n + r] = ns[c];
      anP[(size_t)slice * n + r] = an[c];
    }
  }
}

// merge slices per row, gate by has_pos, fixed-order block tree reduce
__global__ __launch_bounds__(256) void triplet_final(
    const float* __restrict__ psP, const float* __restrict__ nsP,
    const int* __restrict__ anP, const float* __restrict__ maxpos,
    float* __restrict__ out, int n) {
  __shared__ float sL[256];
  __shared__ float sC[256];
  const int tid = threadIdx.x;
  float L = 0.f, C = 0.f;
  for (int r = tid; r < n; r += 256) {
    float p = 0.f, q = 0.f;
    int any = 0;
    for (int s = 0; s < NSLICE; ++s) {
      p   += psP[(size_t)s * n + r];
      q   += nsP[(size_t)s * n + r];
      any |= anP[(size_t)s * n + r];
    }
    const bool haspos = maxpos[r] > -INFINITY;
    if (haspos) { L += p + q; if (any) C += 1.f; }
  }
  sL[tid] = L; sC[tid] = C;
  __syncthreads();
  for (int s = 128; s > 0; s >>= 1) {
    if (tid < s) { sL[tid] += sL[tid + s]; sC[tid] += sC[tid + s]; }
    __syncthreads();
  }
  if (tid == 0) { out[0] = sL[0] / (float)n; out[1] = sC[0]; }
}

extern "C" void kernel_launch(void* const* d_in, const int* in_sizes, int n_in,
                              void* d_out, int out_size, void* d_ws, size_t ws_size,
                              hipStream_t stream) {
  const float* X = (const float*)d_in[0];
  const int* tgt = (const int*)d_in[1];
  const int n = in_sizes[0] / DDIM;        // 8192
  const int nblocks = (n / 128) * NSLICE;  // 512

  float* ws     = (float*)d_ws;
  float* mpP    = ws;                      // NSLICE*n
  float* mnP    = ws + (size_t)NSLICE * n; // NSLICE*n
  float* maxpos = ws + (size_t)2 * NSLICE * n;                 // n
  float* maxneg = maxpos + n;                                  // n
  float* psP    = maxneg + n;                                  // NSLICE*n
  float* nsP    = psP + (size_t)NSLICE * n;                    // NSLICE*n
  int*   anP    = (int*)(nsP + (size_t)NSLICE * n);            // NSLICE*n

  triplet_phase1<<<nblocks, 256, 0, stream>>>(X, tgt, mpP, mnP, n);
  combine_max<<<(n + 255) / 256, 256, 0, stream>>>(mpP, mnP, maxpos, maxneg, n);
  triplet_phase2<<<nblocks, 256, 0, stream>>>(X, tgt, maxpos, maxneg,
                                              psP, nsP, anP, n);
  triplet_final<<<1, 256, 0, stream>>>(psP, nsP, anP, maxpos, (float*)d_out, n);
}